// Memory_73821897884342
// MI455X (gfx1250) — compile-verified
//
#include <hip/hip_runtime.h>
#include <math.h>

#define NTOT (1 << 20)   // number of memory slots
#define EPSV 1e-8f

typedef __attribute__((ext_vector_type(2))) float v2f;
typedef __attribute__((ext_vector_type(8))) float v8f;

// ---------- helpers ----------
__device__ __forceinline__ float waveMax32(float v) {
#pragma unroll
  for (int m = 16; m >= 1; m >>= 1) v = fmaxf(v, __shfl_xor(v, m, 32));
  return v;
}

__device__ __forceinline__ float sel8(const v8f& c, int j) {
  float d = c[0];
  d = (j == 1) ? c[1] : d; d = (j == 2) ? c[2] : d;
  d = (j == 3) ? c[3] : d; d = (j == 4) ? c[4] : d;
  d = (j == 5) ? c[5] : d; d = (j == 6) ? c[6] : d;
  d = (j == 7) ? c[7] : d;
  return d;
}

__device__ __forceinline__ unsigned flipbits(float f) {
  unsigned x = __float_as_uint(f);
  unsigned m = (x & 0x80000000u) ? 0xFFFFFFFFu : 0x80000000u;
  return x ^ m;
}
__device__ __forceinline__ float unflipf(unsigned y) {
  unsigned x = (y & 0x80000000u) ? (y ^ 0x80000000u) : ~y;
  return __uint_as_float(x);
}

// ---------- K0: key norm ----------
__global__ void k0_keynorm(const float* __restrict__ key, float* __restrict__ scal) {
  __shared__ float red[64];
  int t = threadIdx.x;
  float v = key[t];
  red[t] = v * v;
  __syncthreads();
  for (int s = 32; s > 0; s >>= 1) {
    if (t < s) red[t] += red[t + s];
    __syncthreads();
  }
  if (t == 0) scal[0] = fmaxf(sqrtf(red[0]), EPSV);
}

// ---------- K1: WMMA dot + row-norm + score + block max ----------
// K assignment: lanes 0-15 carry K in [0,32) of row (lane&15); lanes 16-31 carry
// K in [32,64). WMMA i consumes pair (2i,2i+1) from the low half and
// (32+2i,32+2i+1) from the high half; B fragments hold the matching key values,
// so the 16-WMMA accumulation equals the full 64-element dot product.
__global__ void k1_score(const float* __restrict__ mem, const float* __restrict__ key,
                         const float* __restrict__ beta, const float* __restrict__ scal,
                         float* __restrict__ score, float* __restrict__ pmax) {
  __shared__ float wred[8];
  const int lane = threadIdx.x & 31;
  const int wid  = threadIdx.x >> 5;
  const int gw   = blockIdx.x * 8 + wid;
  const int tw   = gridDim.x * 8;
  const float keynorm = scal[0];
  const float b = beta[0];
  const int koff = (lane & 16) ? 32 : 0;  // K half this lane covers
  const int r = lane & 15;                // A-fragment row this lane carries

  // preload B fragments (write_key broadcast across all 16 columns)
  float kx[16], ky[16];
#pragma unroll
  for (int i = 0; i < 16; ++i) {
    kx[i] = key[koff + 2 * i];
    ky[i] = key[koff + 2 * i + 1];
  }

  float localMax = -INFINITY;
  const int nTiles = NTOT >> 4;
  for (int t = gw; t < nTiles; t += tw) {
    const float4* rowq =
        reinterpret_cast<const float4*>(mem + ((size_t)((t << 4) + r)) * 64 + koff);
    if (t + tw < nTiles)
      __builtin_prefetch(mem + ((size_t)(((t + tw) << 4) + r)) * 64 + koff, 0, 3);

    // stage the lane's contiguous 128-byte span: 8 independent b128 loads.
    // The sched_barrier keeps the scheduler from sinking each load down to its
    // consumer, so all 8 stay in flight (staged s_wait_loadcnt) instead of
    // serializing one load per WMMA pair.
    float4 q[8];
#pragma unroll
    for (int j = 0; j < 8; ++j) q[j] = rowq[j];
    __builtin_amdgcn_sched_barrier(0);

    v8f c = {0.f, 0.f, 0.f, 0.f, 0.f, 0.f, 0.f, 0.f};
    float ssq = 0.f;
#pragma unroll
    for (int i = 0; i < 16; ++i) {
      const int j = i >> 1;
      float ax = (i & 1) ? q[j].z : q[j].x;
      float ay = (i & 1) ? q[j].w : q[j].y;
      v2f a  = {ax, ay};
      v2f bb = {kx[i], ky[i]};
      c = __builtin_amdgcn_wmma_f32_16x16x4_f32(false, a, false, bb, (short)0, c,
                                                false, false);
      ssq = fmaf(ax, ax, ssq);
      ssq = fmaf(ay, ay, ssq);
    }
    // full sum-of-squares for row (lane&15): combine with partner lane ^16
    float ssqf = ssq + __shfl_xor(ssq, 16, 32);
    const int jr = lane & 7;
    const int needRow = (lane & 16) ? (8 + jr) : jr;  // row whose dot this lane holds
    float ssqn = __shfl(ssqf, needRow, 32);
    if ((lane & 15) < 8) {
      float dot = sel8(c, jr);
      float denom = fmaxf(sqrtf(ssqn), EPSV) * keynorm;
      float s = (dot / denom) * b;
      score[(t << 4) + needRow] = s;
      localMax = fmaxf(localMax, s);
    }
  }
  float wm = waveMax32(localMax);
  if (lane == 0) wred[wid] = wm;
  __syncthreads();
  if (threadIdx.x == 0) {
    float m = wred[0];
#pragma unroll
    for (int i = 1; i < 8; ++i) m = fmaxf(m, wred[i]);
    pmax[blockIdx.x] = m;
  }
}

// ---------- reductions ----------
__global__ void k_reduce_max(const float* __restrict__ in, int n, float* __restrict__ out) {
  __shared__ float r[1024];
  float m = -INFINITY;
  for (int i = threadIdx.x; i < n; i += blockDim.x) m = fmaxf(m, in[i]);
  r[threadIdx.x] = m;
  __syncthreads();
  for (int s = blockDim.x >> 1; s > 0; s >>= 1) {
    if ((int)threadIdx.x < s) r[threadIdx.x] = fmaxf(r[threadIdx.x], r[threadIdx.x + s]);
    __syncthreads();
  }
  if (threadIdx.x == 0) out[0] = r[0];
}

__global__ void k_reduce_sum(const float* __restrict__ in, int n, float* __restrict__ out) {
  __shared__ float r[1024];
  float a = 0.f;
  for (int i = threadIdx.x; i < n; i += blockDim.x) a += in[i];
  r[threadIdx.x] = a;
  __syncthreads();
  for (int s = blockDim.x >> 1; s > 0; s >>= 1) {
    if ((int)threadIdx.x < s) r[threadIdx.x] += r[threadIdx.x + s];
    __syncthreads();
  }
  if (threadIdx.x == 0) out[0] = r[0];
}

// ---------- exp pass (softmax numerator) + partial sums ----------
__global__ void k_exp(float* __restrict__ A, const float* __restrict__ scal,
                      float* __restrict__ psum) {
  __shared__ float r[256];
  const float M = scal[1];
  float acc = 0.f;
  for (int n = blockIdx.x * blockDim.x + threadIdx.x; n < NTOT;
       n += gridDim.x * blockDim.x) {
    float e = expf(A[n] - M);
    A[n] = e;
    acc += e;
  }
  r[threadIdx.x] = acc;
  __syncthreads();
  for (int s = 128; s > 0; s >>= 1) {
    if ((int)threadIdx.x < s) r[threadIdx.x] += r[threadIdx.x + s];
    __syncthreads();
  }
  if (threadIdx.x == 0) psum[blockIdx.x] = r[0];
}

// ---------- usage ----------
__global__ void k2_usage(const float* __restrict__ rw, const float* __restrict__ fg,
                         const float* __restrict__ pu, const float* __restrict__ pww,
                         float* __restrict__ usage) {
  int n = blockIdx.x * blockDim.x + threadIdx.x;
  const float4* r4 = reinterpret_cast<const float4*>(rw + (size_t)n * 8);
  float4 a = r4[0], b = r4[1];
  float ret = (1.f - a.x * fg[0]) * (1.f - a.y * fg[1]) * (1.f - a.z * fg[2]) *
              (1.f - a.w * fg[3]) * (1.f - b.x * fg[4]) * (1.f - b.y * fg[5]) *
              (1.f - b.z * fg[6]) * (1.f - b.w * fg[7]);
  float u = pu[n], w = pww[n];
  usage[n] = (u + w - u * w) * ret;
}

// ---------- radix sort: init keys ----------
__global__ void k_init_keys(const float* __restrict__ usage, unsigned* __restrict__ keys) {
  int n = blockIdx.x * blockDim.x + threadIdx.x;
  keys[n] = flipbits(usage[n]);
}

// ---------- radix sort: per-block histogram (digit-major output) ----------
__global__ void k_hist(const unsigned* __restrict__ keys, unsigned* __restrict__ hist,
                       int shift) {
  __shared__ unsigned h[256];
  h[threadIdx.x] = 0;
  __syncthreads();
  unsigned key = keys[blockIdx.x * 256 + threadIdx.x];
  atomicAdd(&h[(key >> shift) & 255u], 1u);
  __syncthreads();
  hist[threadIdx.x * gridDim.x + blockIdx.x] = h[threadIdx.x];
}

// ---------- radix sort: single-block exclusive scan of 256*4096 counts ----------
__global__ void k_scan_hist(unsigned* __restrict__ h) {
  __shared__ unsigned sums[1024];
  const int t = threadIdx.x;
  const int TOTAL = 256 * 4096;
  const int SEG = TOTAL / 1024;  // 1024
  int s0 = t * SEG;
  unsigned run = 0;
  for (int i = 0; i < SEG; ++i) {
    unsigned v = h[s0 + i];
    h[s0 + i] = run;
    run += v;
  }
  sums[t] = run;
  __syncthreads();
  for (int off = 1; off < 1024; off <<= 1) {
    unsigned v = (t >= off) ? sums[t - off] : 0u;
    __syncthreads();
    sums[t] += v;
    __syncthreads();
  }
  unsigned base = (t == 0) ? 0u : sums[t - 1];
  for (int i = 0; i < SEG; ++i) h[s0 + i] += base;
}

// ---------- radix sort: stable scatter (ballot-ranked, no global atomics) ----------
__global__ void k_scatter(const unsigned* __restrict__ kin, const unsigned* __restrict__ iin,
                          unsigned* __restrict__ kout, unsigned* __restrict__ iout,
                          const unsigned* __restrict__ histScan, int shift, int pass0) {
  __shared__ unsigned waveCnt[8][256];
  const int lane = threadIdx.x & 31;
  const int wid  = threadIdx.x >> 5;
  unsigned gid = blockIdx.x * 256 + threadIdx.x;
  unsigned key = kin[gid];
  unsigned idx = pass0 ? gid : iin[gid];
  unsigned d = (key >> shift) & 255u;

  // lanes within this wave sharing my digit
  unsigned peers = 0xFFFFFFFFu;
#pragma unroll
  for (int b = 0; b < 8; ++b) {
    unsigned bit = (d >> b) & 1u;
    unsigned vote = (unsigned)__ballot(bit != 0u);
    peers &= bit ? vote : ~vote;
  }
  unsigned below = (lane == 0) ? 0u : (0xFFFFFFFFu >> (32 - lane));
  unsigned rank = __popc(peers & below);
  unsigned cnt  = __popc(peers);

  unsigned* flat = &waveCnt[0][0];
  for (int i = threadIdx.x; i < 8 * 256; i += 256) flat[i] = 0u;
  __syncthreads();
  if (rank == 0) waveCnt[wid][d] = cnt;
  __syncthreads();
  // per-digit exclusive scan over the 8 waves; thread t owns digit t
  {
    unsigned run = 0;
    for (int w = 0; w < 8; ++w) {
      unsigned v = waveCnt[w][threadIdx.x];
      waveCnt[w][threadIdx.x] = run;
      run += v;
    }
  }
  __syncthreads();
  unsigned pos = histScan[d * gridDim.x + blockIdx.x] + waveCnt[wid][d] + rank;
  kout[pos] = key;
  iout[pos] = idx;
}

// ---------- exclusive cumprod: per-block scan ----------
__global__ void k_cp1(const unsigned* __restrict__ keys, float* __restrict__ locExcl,
                      float* __restrict__ bprod) {
  __shared__ float sc[256];
  int t = threadIdx.x;
  int gid = blockIdx.x * 256 + t;
  float u = unflipf(keys[gid]);
  sc[t] = u;
  __syncthreads();
  for (int off = 1; off < 256; off <<= 1) {
    float v = (t >= off) ? sc[t - off] : 1.f;
    __syncthreads();
    sc[t] *= v;
    __syncthreads();
  }
  locExcl[gid] = (t == 0) ? 1.f : sc[t - 1];
  if (t == 255) bprod[blockIdx.x] = sc[255];
}

// ---------- exclusive cumprod: scan of 4096 block products ----------
__global__ void k_cp2(const float* __restrict__ bp, float* __restrict__ bpre) {
  __shared__ float s[1024];
  int t = threadIdx.x;
  float v0 = bp[t * 4], v1 = bp[t * 4 + 1], v2 = bp[t * 4 + 2], v3 = bp[t * 4 + 3];
  s[t] = v0 * v1 * v2 * v3;
  __syncthreads();
  for (int off = 1; off < 1024; off <<= 1) {
    float v = (t >= off) ? s[t - off] : 1.f;
    __syncthreads();
    s[t] *= v;
    __syncthreads();
  }
  float base = (t == 0) ? 1.f : s[t - 1];
  bpre[t * 4] = base; base *= v0;
  bpre[t * 4 + 1] = base; base *= v1;
  bpre[t * 4 + 2] = base; base *= v2;
  bpre[t * 4 + 3] = base;
}

// ---------- allocation scatter ----------
__global__ void k_alloc(const unsigned* __restrict__ keys, const unsigned* __restrict__ idx,
                        const float* __restrict__ locExcl, const float* __restrict__ bpre,
                        float* __restrict__ alloc) {
  int gid = blockIdx.x * 256 + threadIdx.x;
  float u = unflipf(keys[gid]);
  float ecp = bpre[blockIdx.x] * locExcl[gid];
  alloc[idx[gid]] = (1.f - u) * ecp;
}

// ---------- final gating + partial sums of ww ----------
__global__ void k_final(const float* __restrict__ e, const float* __restrict__ alloc,
                        const float* __restrict__ scal, const float* __restrict__ agp,
                        const float* __restrict__ wgp, float* __restrict__ out,
                        float* __restrict__ psum) {
  __shared__ float r[256];
  const float S = scal[2];
  const float ag = agp[0], wg = wgp[0];
  float acc = 0.f;
  for (int n = blockIdx.x * blockDim.x + threadIdx.x; n < NTOT;
       n += gridDim.x * blockDim.x) {
    float cw = e[n] / S;
    float ww = wg * (ag * alloc[n] + (1.f - ag) * cw);
    out[n] = ww;
    acc += ww;
  }
  r[threadIdx.x] = acc;
  __syncthreads();
  for (int s = 128; s > 0; s >>= 1) {
    if ((int)threadIdx.x < s) r[threadIdx.x] += r[threadIdx.x + s];
    __syncthreads();
  }
  if (threadIdx.x == 0) psum[blockIdx.x] = r[0];
}

// ---------- precedence update ----------
__global__ void k_prec(const float* __restrict__ prec, const float* __restrict__ scal,
                       float* __restrict__ out) {
  int n = blockIdx.x * blockDim.x + threadIdx.x;
  float s = scal[3];
  out[2 * NTOT + n] = (1.f - s) * prec[n] + out[n];
}

// ---------- host launch ----------
extern "C" void kernel_launch(void* const* d_in, const int* in_sizes, int n_in,
                              void* d_out, int out_size, void* d_ws, size_t ws_size,
                              hipStream_t stream) {
  (void)in_sizes; (void)n_in; (void)out_size; (void)ws_size;
  const float* mem  = (const float*)d_in[0];
  const float* key  = (const float*)d_in[1];
  const float* beta = (const float*)d_in[2];
  const float* fg   = (const float*)d_in[3];
  const float* rw   = (const float*)d_in[4];
  const float* pu   = (const float*)d_in[5];
  const float* pww  = (const float*)d_in[6];
  const float* ag   = (const float*)d_in[7];
  const float* wg   = (const float*)d_in[8];
  const float* prec = (const float*)d_in[9];
  float* out   = (float*)d_out;
  float* usage = out + NTOT;           // outputs: [ww | usage | new_precedence]

  char* w = (char*)d_ws;
  float*    A    = (float*)(w);                         // scores -> exp values (4MB)
  float*    B    = (float*)(w + ((size_t)4  << 20));    // allocation weighting (4MB)
  unsigned* C    = (unsigned*)(w + ((size_t)8  << 20)); // keys ping
  unsigned* D    = (unsigned*)(w + ((size_t)12 << 20)); // keys pong
  unsigned* E    = (unsigned*)(w + ((size_t)16 << 20)); // idx ping
  unsigned* F    = (unsigned*)(w + ((size_t)20 << 20)); // idx pong
  unsigned* G    = (unsigned*)(w + ((size_t)24 << 20)); // digit-major histogram
  float*    H    = (float*)(w + ((size_t)28 << 20));    // per-block excl cumprod
  float*    Pmax = (float*)(w + ((size_t)32 << 20));
  float*    Psum = Pmax + 2048;
  float*    CPb  = Psum + 2048;
  float*    CPp  = CPb + 4096;
  float*    Pww  = CPp + 4096;
  float*    SC   = Pww + 2048;  // [keynorm, M, S, sum_ww]

  k0_keynorm<<<1, 64, 0, stream>>>(key, SC);
  k2_usage<<<4096, 256, 0, stream>>>(rw, fg, pu, pww, usage);
  k1_score<<<2048, 256, 0, stream>>>(mem, key, beta, SC, A, Pmax);
  k_reduce_max<<<1, 1024, 0, stream>>>(Pmax, 2048, SC + 1);
  k_exp<<<2048, 256, 0, stream>>>(A, SC, Psum);
  k_reduce_sum<<<1, 1024, 0, stream>>>(Psum, 2048, SC + 2);

  // radix sort usage ascending (stable LSD, 4 x 8-bit)
  k_init_keys<<<4096, 256, 0, stream>>>(usage, C);
  k_hist<<<4096, 256, 0, stream>>>(C, G, 0);
  k_scan_hist<<<1, 1024, 0, stream>>>(G);
  k_scatter<<<4096, 256, 0, stream>>>(C, E, D, E, G, 0, 1);
  k_hist<<<4096, 256, 0, stream>>>(D, G, 8);
  k_scan_hist<<<1, 1024, 0, stream>>>(G);
  k_scatter<<<4096, 256, 0, stream>>>(D, E, C, F, G, 8, 0);
  k_hist<<<4096, 256, 0, stream>>>(C, G, 16);
  k_scan_hist<<<1, 1024, 0, stream>>>(G);
  k_scatter<<<4096, 256, 0, stream>>>(C, F, D, E, G, 16, 0);
  k_hist<<<4096, 256, 0, stream>>>(D, G, 24);
  k_scan_hist<<<1, 1024, 0, stream>>>(G);
  k_scatter<<<4096, 256, 0, stream>>>(D, E, C, F, G, 24, 0);
  // sorted keys in C, original indices in F

  k_cp1<<<4096, 256, 0, stream>>>(C, H, CPb);
  k_cp2<<<1, 1024, 0, stream>>>(CPb, CPp);
  k_alloc<<<4096, 256, 0, stream>>>(C, F, H, CPp, B);

  k_final<<<2048, 256, 0, stream>>>(A, B, SC, ag, wg, out, Pww);
  k_reduce_sum<<<1, 1024, 0, stream>>>(Pww, 2048, SC + 3);
  k_prec<<<4096, 256, 0, stream>>>(prec, SC, out);
}